// GCN_48850958024707
// MI455X (gfx1250) — compile-verified
//
#include <hip/hip_runtime.h>

#define NND   100000
#define NEDG  1600000
#define FDIM  128
#define NCLS  40

typedef __attribute__((ext_vector_type(2))) float v2f;
typedef __attribute__((ext_vector_type(8))) float v8f;

__device__ __forceinline__ void atomic_add_f32(float* p, float v) {
  // relaxed, device scope -> lowers to global_atomic_add_f32 (no return)
  __hip_atomic_fetch_add(p, v, __ATOMIC_RELAXED, __HIP_MEMORY_SCOPE_AGENT);
}

// ---------------- degree / normalization ----------------
__global__ void k_set1(float* __restrict__ deg, int n) {
  int i = blockIdx.x * blockDim.x + threadIdx.x;
  if (i < n) deg[i] = 1.0f;
}

__global__ void k_count(const int* __restrict__ dst, float* __restrict__ deg, int ne) {
  int i = blockIdx.x * blockDim.x + threadIdx.x;
  if (i < ne) atomic_add_f32(&deg[dst[i]], 1.0f);
}

__global__ void k_rsqrt(float* __restrict__ d, int n) {
  int i = blockIdx.x * blockDim.x + threadIdx.x;
  if (i < n) d[i] = rsqrtf(d[i]);
}

// ---------------- fp32 WMMA GEMM: out[nrows x NCOL] = A[nrows x 128] * W[128 x NCOL] (+bias) ----------------
// One wave computes a 16-row strip across all column tiles.
// A-operand layout (16x4 f32): lanes 0-15 hold M=lane, K={k0,k0+1}; lanes 16-31 K={k0+2,k0+3}.
// B-operand layout (4x16 f32): lanes 0-15 hold N=lane, K={0,1}; lanes 16-31 K={2,3}.
template <int NCOL>
__global__ void k_gemm_wmma(const float* __restrict__ A, const float* __restrict__ W,
                            const float* __restrict__ bias, float* __restrict__ out,
                            int nrows) {
  const int wid  = threadIdx.x >> 5;
  const int lane = threadIdx.x & 31;
  const int rowTile = blockIdx.x * (blockDim.x >> 5) + wid;
  if (rowTile * 16 >= nrows) return;   // wave-uniform: EXEC stays all-ones for WMMA

  const int m  = lane & 15;            // row within tile (A) / col within tile (B,C)
  const int hf = lane >> 4;            // half selector
  const int k0 = hf * 2;

  constexpr int NT = (NCOL + 15) / 16;
  v8f c[NT] = {};

  const float* arow = A + (size_t)(rowTile * 16 + m) * FDIM;

  for (int k = 0; k < FDIM; k += 4) {
    v2f a;
    a.x = arow[k + k0];
    a.y = arow[k + k0 + 1];
#pragma unroll
    for (int t = 0; t < NT; ++t) {
      const int col = t * 16 + m;
      v2f b;
      if ((NCOL % 16 == 0) || (col < NCOL)) {
        b.x = W[(size_t)(k + k0)     * NCOL + col];
        b.y = W[(size_t)(k + k0 + 1) * NCOL + col];
      } else {
        b.x = 0.0f; b.y = 0.0f;
      }
      c[t] = __builtin_amdgcn_wmma_f32_16x16x4_f32(
          /*neg_a=*/false, a, /*neg_b=*/false, b,
          /*c_mod=*/(short)0, c[t], /*reuse_a=*/false, /*reuse_b=*/false);
    }
  }

  // C/D layout: VGPR r -> row (r + hf*8), col = lane&15 within tile
#pragma unroll
  for (int t = 0; t < NT; ++t) {
    const int col = t * 16 + m;
    if ((NCOL % 16 != 0) && (col >= NCOL)) continue;
    const float bv = (bias != nullptr) ? bias[col] : 0.0f;
#pragma unroll
    for (int r = 0; r < 8; ++r) {
      const int row = rowTile * 16 + r + hf * 8;
      out[(size_t)row * NCOL + col] = c[t][r] + bv;
    }
  }
}

// ---------------- self-loop init: acc[i,:] = h[i,:] * dinv[i]^2 ----------------
__global__ void k_self(const float* __restrict__ h, const float* __restrict__ dinv,
                       float* __restrict__ acc, int n) {
  int idx = blockIdx.x * blockDim.x + threadIdx.x;  // n*32 threads, float4 each
  int node = idx >> 5;
  int q    = idx & 31;
  if (node >= n) return;
  float di = dinv[node];
  float s  = di * di;
  float4 v = ((const float4*)(h + (size_t)node * FDIM))[q];
  float4 o = make_float4(v.x * s, v.y * s, v.z * s, v.w * s);
  ((float4*)(acc + (size_t)node * FDIM))[q] = o;
}

// ---------------- edge scatter: acc[dst,:] += h[src,:] * dinv[src]*dinv[dst] ----------------
// One wave per edge: 32 lanes x float4 = 128 features.
__global__ void k_edge(const int* __restrict__ src, const int* __restrict__ dst,
                       const float* __restrict__ dinv, const float* __restrict__ h,
                       float* __restrict__ acc, int ne) {
  int wid  = threadIdx.x >> 5;
  int lane = threadIdx.x & 31;
  int e = blockIdx.x * (blockDim.x >> 5) + wid;
  if (e >= ne) return;
  int s = src[e];
  int d = dst[e];
  float coeff = dinv[s] * dinv[d];
  float4 v = ((const float4*)(h + (size_t)s * FDIM))[lane];
  float* p = acc + (size_t)d * FDIM + lane * 4;
  atomic_add_f32(p + 0, v.x * coeff);
  atomic_add_f32(p + 1, v.y * coeff);
  atomic_add_f32(p + 2, v.z * coeff);
  atomic_add_f32(p + 3, v.w * coeff);
}

// ---------------- bias + ReLU ----------------
__global__ void k_bias_relu(const float* __restrict__ acc, const float* __restrict__ b,
                            float* __restrict__ out, int n) {
  int idx = blockIdx.x * blockDim.x + threadIdx.x;
  int node = idx >> 5;
  int q    = idx & 31;
  if (node >= n) return;
  float4 v  = ((const float4*)(acc + (size_t)node * FDIM))[q];
  float4 bb = ((const float4*)b)[q];
  float4 o  = make_float4(fmaxf(v.x + bb.x, 0.0f), fmaxf(v.y + bb.y, 0.0f),
                          fmaxf(v.z + bb.z, 0.0f), fmaxf(v.w + bb.w, 0.0f));
  ((float4*)(out + (size_t)node * FDIM))[q] = o;
}

extern "C" void kernel_launch(void* const* d_in, const int* in_sizes, int n_in,
                              void* d_out, int out_size, void* d_ws, size_t ws_size,
                              hipStream_t stream) {
  const float* x    = (const float*)d_in[0];
  const int*   ei   = (const int*)  d_in[1];
  const float* W1   = (const float*)d_in[2];
  const float* b1   = (const float*)d_in[3];
  const float* W2   = (const float*)d_in[4];
  const float* b2   = (const float*)d_in[5];
  const float* Wout = (const float*)d_in[6];
  const float* bout = (const float*)d_in[7];
  float* out = (float*)d_out;

  const int* src = ei;             // edge_index row 0
  const int* dst = ei + NEDG;      // edge_index row 1

  // workspace layout (floats): dinv | h (N*128) | agg (N*128); ~103 MB total
  float* ws   = (float*)d_ws;
  float* dinv = ws;
  float* h    = ws + (1u << 17);                       // 512 KB offset (16B aligned)
  float* agg  = h + (size_t)NND * FDIM;                // 16B aligned (N*128*4 % 16 == 0)

  const dim3 B(256);
  const int nodeBlk  = (NND + 255) / 256;
  const int edgeBlk1 = (NEDG + 255) / 256;
  const int rowTiles = NND / 16;                       // 6250 (exact)
  const int gemmBlk  = (rowTiles + 7) / 8;             // 8 waves per block
  const int featBlk  = (NND * 32 + 255) / 256;         // node x float4 threads
  const int edgeBlkW = (NEDG + 7) / 8;                 // 1 wave per edge

  // normalization: deg -> dinv (once, shared by both layers)
  k_set1 <<<nodeBlk,  B, 0, stream>>>(dinv, NND);
  k_count<<<edgeBlk1, B, 0, stream>>>(dst, dinv, NEDG);
  k_rsqrt<<<nodeBlk,  B, 0, stream>>>(dinv, NND);

  // ---- layer 1: h = x@W1 ; agg = scatter(h) ; h = relu(agg + b1) ----
  k_gemm_wmma<FDIM><<<gemmBlk, B, 0, stream>>>(x, W1, nullptr, h, NND);
  k_self     <<<featBlk, B, 0, stream>>>(h, dinv, agg, NND);
  k_edge     <<<edgeBlkW, B, 0, stream>>>(src, dst, dinv, h, agg, NEDG);
  k_bias_relu<<<featBlk, B, 0, stream>>>(agg, b1, h, NND);

  // ---- layer 2 (ping-pong: gemm -> agg, accumulate -> h) ----
  k_gemm_wmma<FDIM><<<gemmBlk, B, 0, stream>>>(h, W2, nullptr, agg, NND);
  k_self     <<<featBlk, B, 0, stream>>>(agg, dinv, h, NND);
  k_edge     <<<edgeBlkW, B, 0, stream>>>(src, dst, dinv, agg, h, NEDG);
  k_bias_relu<<<featBlk, B, 0, stream>>>(h, b2, h, NND);

  // ---- output projection: out = h@Wout + bout (40 cols, masked tiles) ----
  k_gemm_wmma<NCLS><<<gemmBlk, B, 0, stream>>>(h, Wout, bout, out, NND);
}